// MuellerMatrixPyramid_28802050687750
// MI455X (gfx1250) — compile-verified
//
#include <hip/hip_runtime.h>
#include <hip/hip_bf16.h>

// MI455X (gfx1250) implementation. Memory-bound fp32 pipeline; the per-pixel
// 4x4 matrix product M = inv(A)@I @ inv(W) is batched 4-pixels-at-a-time onto
// V_WMMA_F32_16X16X4_F32 via block-diagonal packing through LDS.
// Round 3: all accumulator extracts use constant indices (the previous
// hi?d[r+4]:d[r] select became a divergent dynamic extractelement -> 7-deep
// cndmask trees per element). Selection now lives purely in the ds_store
// addresses: two streams (d[0..3], d[4..7]); the non-selected stream collapses
// onto the lane's private LDS pad slot via step=0.

#define BATCH 8
#define H0 512
#define W0 512
#define H1 256
#define W1 256

typedef __attribute__((ext_vector_type(2))) float v2f;
typedef __attribute__((ext_vector_type(8))) float v8f;

// Per-block LDS staging. 8 waves x 32 lanes x 16 floats, inner dim padded to
// 17: stride-17 lane-major access is bank-conflict-free on 64 banks, and the
// pad element [L][16] doubles as the dump target for the branch-free scatter.
// XD holds X = inv(A)@I entering the WMMA phase and is overwritten in-place by
// D = X@Winv (safe: pixel p's X is read in the same group iteration that
// writes D[p], and same-wave LDS instructions execute in order).
struct WaveLDS {
  float XD[8][32][17];
  float W[8][32][17];
};

// Gauss-Jordan solve of a 4x4 system with 4 RHS: x := inv(a) @ x (a destroyed).
// Calibration matrices are I + 0.05*N(0,1): diagonal pivots are safe.
__device__ __forceinline__ void gauss_solve4(float* a, float* x) {
#pragma unroll
  for (int c = 0; c < 4; ++c) {
    const float inv = 1.0f / a[c * 4 + c];
#pragma unroll
    for (int k = 0; k < 4; ++k) {
      a[c * 4 + k] *= inv;
      x[c * 4 + k] *= inv;
    }
#pragma unroll
    for (int r = 0; r < 4; ++r) {
      if (r == c) continue;
      const float f = a[r * 4 + c];
#pragma unroll
      for (int k = 0; k < 4; ++k) {
        a[r * 4 + k] -= f * a[c * 4 + k];
        x[r * 4 + k] -= f * x[c * 4 + k];
      }
    }
  }
}

// Wave-cooperative: each of the 32 lanes owns one pixel's 4x4 X and Winv.
// One V_WMMA_F32_16X16X4_F32 multiplies 4 pixels (block-diagonal packing):
//   A(16x4)  rows 4q+i = X_q row i       (VGPR0: K=0/2 by lane half, VGPR1: K=1/3)
//   B(4x16)  cols 4q+j = Winv_q col j    (same K split across lane halves)
//   D(16x16) diagonal blocks D[4q+i][4q+j] = (X_q @ Winv_q)[i][j]
// 8 WMMAs cover the wave's 32 pixels. EXEC is all-1s (grids divide evenly,
// no divergent control flow around the WMMAs).
__device__ __forceinline__ void mm_multiply_wmma(WaveLDS& s, const float* X,
                                                 const float* Wi, float* M) {
  const int w = threadIdx.x >> 5;
  const int L = threadIdx.x & 31;
  float* lds = &s.XD[0][0][0];
  const int wbase = w * (32 * 17);

#pragma unroll
  for (int i = 0; i < 16; ++i) {
    s.XD[w][L][i] = X[i];
    s.W[w][L][i] = Wi[i];
  }
  // Same-wave LDS ops are in-order; only compiler reordering must be fenced.
  __builtin_amdgcn_wave_barrier();

  const int mn   = L & 15;    // row index (A role) == col index (B role)
  const int half = L >> 4;    // lane half selects K = {0,1} vs {2,3}
  const int q    = mn >> 2;   // pixel within 4-pixel group
  const int ij   = mn & 3;    // row i (A) / col j (B) within the 4x4

  // Lane holds D rows m = r + 8*half (r=0..7), column n = mn. The in-block
  // elements (m>>2 == q) exist only if (q>>1)==half; they are d[0..3] when
  // q is even, d[4..7] when q is odd. Store both halves unconditionally with
  // constant-index extracts; route the unused half to the pad slot (step=0).
  const bool contrib = (q >> 1) == half;
  const bool hi      = (q & 1) != 0;
  const bool useLo   = contrib && !hi;  // d[0..3] are the real block
  const bool useHi   = contrib && hi;   // d[4..7] are the real block
  const int dummyOff = wbase + L * 17 + 16;
  const int stepLo   = useLo ? 4 : 0;
  const int stepHi   = useHi ? 4 : 0;

#pragma unroll
  for (int g = 0; g < 8; ++g) {
    const int p = g * 4 + q;  // lane's pixel for this group
    v2f a, b;
    a.x = s.XD[w][p][ij * 4 + half * 2 + 0];     // X_p[i][k]
    a.y = s.XD[w][p][ij * 4 + half * 2 + 1];
    b.x = s.W[w][p][(half * 2 + 0) * 4 + ij];    // Winv_p[k][j]
    b.y = s.W[w][p][(half * 2 + 1) * 4 + ij];
    v8f cz = {0.f, 0.f, 0.f, 0.f, 0.f, 0.f, 0.f, 0.f};
    v8f d = __builtin_amdgcn_wmma_f32_16x16x4_f32(
        /*neg_a=*/false, a, /*neg_b=*/false, b,
        /*c_mod=*/(short)0, cz, /*reuse_a=*/false, /*reuse_b=*/false);

    const int realBase = wbase + p * 17 + ij;
    const int baseLo = useLo ? realBase : dummyOff;
    const int baseHi = useHi ? realBase : dummyOff;
#pragma unroll
    for (int r = 0; r < 4; ++r) {
      lds[baseLo + r * stepLo] = d[r];      // D[p][r*4 + ij] or pad dump
      lds[baseHi + r * stepHi] = d[r + 4];  // D[p][r*4 + ij] or pad dump
    }
  }
  __builtin_amdgcn_wave_barrier();
#pragma unroll
  for (int i = 0; i < 16; ++i) M[i] = s.XD[w][L][i];
}

// v48 = {I[16], A[16], W[16]} for one pixel -> out17 = {mean(I), M/M00 [16]}
__device__ __forceinline__ void mm_features(WaveLDS& s, const float* v48,
                                            float* out17) {
  float Xm[16], Am[16];
  float inten = 0.0f;
#pragma unroll
  for (int i = 0; i < 16; ++i) {
    Xm[i] = v48[i];
    inten += v48[i];
    Am[i] = v48[16 + i];
  }
  inten *= 0.0625f;
  gauss_solve4(Am, Xm);  // Xm := inv(A) @ I

  float Wm[16], Wi[16];
#pragma unroll
  for (int i = 0; i < 16; ++i) {
    Wm[i] = v48[32 + i];
    Wi[i] = ((i & 3) == (i >> 2)) ? 1.0f : 0.0f;  // identity
  }
  gauss_solve4(Wm, Wi);  // Wi := inv(W)

  float M[16];
  mm_multiply_wmma(s, Xm, Wi, M);

  const float n = 1.0f / M[0];
  out17[0] = inten;
#pragma unroll
  for (int i = 0; i < 16; ++i) out17[1 + i] = M[i] * n;
}

// Level 0: full-res features -> out channels [0, 17)
__global__ void __launch_bounds__(256) k_level0(const float* __restrict__ x,
                                                float* __restrict__ out) {
  __shared__ WaveLDS s;
  const int tid = blockIdx.x * 256 + threadIdx.x;
  const int xw = tid & (W0 - 1);
  const int yh = (tid >> 9) & (H0 - 1);
  const int b = tid >> 18;
  const size_t plane = (size_t)H0 * W0;

  const float* px = x + (size_t)b * 48 * plane + (size_t)yh * W0 + xw;
  float v[48];
#pragma unroll
  for (int c = 0; c < 48; ++c) v[c] = px[(size_t)c * plane];

  float f[17];
  mm_features(s, v, f);

  float* po = out + (size_t)b * 34 * plane + (size_t)yh * W0 + xw;
#pragma unroll
  for (int c = 0; c < 17; ++c) po[(size_t)c * plane] = f[c];
}

// Level 1: fused 2x2 maxpool + features at half res -> feat [B,17,256,256]
__global__ void __launch_bounds__(256) k_level1(const float* __restrict__ x,
                                                float* __restrict__ feat) {
  __shared__ WaveLDS s;
  const int tid = blockIdx.x * 256 + threadIdx.x;
  const int xw = tid & (W1 - 1);
  const int yh = (tid >> 8) & (H1 - 1);
  const int b = tid >> 16;
  const size_t plane = (size_t)H0 * W0;
  const size_t plane1 = (size_t)H1 * W1;

  const float* px =
      x + (size_t)b * 48 * plane + (size_t)(2 * yh) * W0 + (size_t)(2 * xw);
  float v[48];
#pragma unroll
  for (int c = 0; c < 48; ++c) {
    const float* p = px + (size_t)c * plane;
    const float2 r0 = *reinterpret_cast<const float2*>(p);
    const float2 r1 = *reinterpret_cast<const float2*>(p + W0);
    v[c] = fmaxf(fmaxf(r0.x, r0.y), fmaxf(r1.x, r1.y));
  }

  float f[17];
  mm_features(s, v, f);

  float* po = feat + (size_t)b * 17 * plane1 + (size_t)yh * W1 + xw;
#pragma unroll
  for (int c = 0; c < 17; ++c) po[(size_t)c * plane1] = f[c];
}

// Bilinear align_corners=True upsample of feat -> out channels [17, 34)
__global__ void __launch_bounds__(256) k_upsample(const float* __restrict__ feat,
                                                  float* __restrict__ out) {
  const int tid = blockIdx.x * 256 + threadIdx.x;
  const int xw = tid & (W0 - 1);
  int t = tid >> 9;
  const int yh = t & (H0 - 1);
  t >>= 9;
  const int c = t % 17;
  const int b = t / 17;

  const float sc = (float)(H1 - 1) / (float)(H0 - 1);  // 255/511
  const float ysf = yh * sc;
  const float xsf = xw * sc;
  const int y0 = (int)ysf;
  const int x0 = (int)xsf;
  const float fy = ysf - (float)y0;
  const float fx = xsf - (float)x0;
  int y1 = y0 + 1; if (y1 > H1 - 1) y1 = H1 - 1;
  int x1 = x0 + 1; if (x1 > W1 - 1) x1 = W1 - 1;

  const float* p = feat + (size_t)(b * 17 + c) * (H1 * W1);
  const float v00 = p[y0 * W1 + x0];
  const float v01 = p[y0 * W1 + x1];
  const float v10 = p[y1 * W1 + x0];
  const float v11 = p[y1 * W1 + x1];
  const float r0 = v00 + (v01 - v00) * fx;
  const float r1 = v10 + (v11 - v10) * fx;

  out[(((size_t)b * 34 + 17 + c) * H0 + yh) * W0 + xw] = r0 + (r1 - r0) * fy;
}

extern "C" void kernel_launch(void* const* d_in, const int* in_sizes, int n_in,
                              void* d_out, int out_size, void* d_ws,
                              size_t ws_size, hipStream_t stream) {
  (void)in_sizes; (void)n_in; (void)out_size; (void)ws_size;
  const float* x = (const float*)d_in[0];
  float* out = (float*)d_out;
  float* feat1 = (float*)d_ws;  // [8,17,256,256] fp32 = ~34 MB scratch

  k_level0<<<(BATCH * H0 * W0) / 256, 256, 0, stream>>>(x, out);
  k_level1<<<(BATCH * H1 * W1) / 256, 256, 0, stream>>>(x, feat1);
  k_upsample<<<(BATCH * 17 * H0 * W0) / 256, 256, 0, stream>>>(feat1, out);
}